// NearestEmbedModule_31911607009948
// MI455X (gfx1250) — compile-verified
//
#include <hip/hip_runtime.h>

typedef __attribute__((ext_vector_type(2))) float        v2f;
typedef __attribute__((ext_vector_type(8))) float        v8f;
typedef __attribute__((ext_vector_type(4))) unsigned int u32x4;
typedef __attribute__((ext_vector_type(8))) int          i32x8;
typedef __attribute__((ext_vector_type(4))) int          i32x4;

#define D_DIM 256      // embedding dim (reduction dim of GEMM)
#define K_DIM 1024     // number of embeddings (columns)
#define NTILES (K_DIM / 16)
#define ROWS_PER_WG 128
#define NWAVES 8       // 256 threads / 32

#if defined(__has_builtin)
#if __has_builtin(__builtin_amdgcn_tensor_load_to_lds)
#define USE_TDM 1
#endif
#endif

// ---------------------------------------------------------------------------
// Kernel 1: e2[k] = sum_d W[d][k]^2   (coalesced over k across lanes)
// ---------------------------------------------------------------------------
__global__ __launch_bounds__(256)
void vq_e2_kernel(const float* __restrict__ w, float* __restrict__ e2) {
  __shared__ float red[256];
  const int t = threadIdx.x;
  const int k = blockIdx.x * 64 + (t & 63);
  const int dc = t >> 6;                  // 4 d-chunks of 64
  float acc = 0.0f;
  #pragma unroll 4
  for (int d = dc * 64; d < dc * 64 + 64; ++d) {
    float v = w[(size_t)d * K_DIM + k];
    acc += v * v;
  }
  red[t] = acc;
  __syncthreads();
  if (t < 64) {
    e2[blockIdx.x * 64 + t] = red[t] + red[t + 64] + red[t + 128] + red[t + 192];
  }
}

#ifdef USE_TDM
// Issue a TDM DMA of one W panel (256 rows x 16 cols, f32) into LDS.
// 2D descriptor: tile_dim0=16 (contiguous n), tile_dim1=256 (d rows),
// tensor_dim0=1024, dim0_stride=1024 elements, data_size=4B.
__device__ __forceinline__ void tdm_load_panel(const float* gsrc, unsigned lds_off) {
  const unsigned long long ga = (unsigned long long)(uintptr_t)gsrc;
  u32x4 g0;
  g0.x = 1u;                                            // count=1 (valid user D#)
  g0.y = lds_off;                                       // LDS byte address
  g0.z = (unsigned)(ga & 0xffffffffu);                  // global_addr[31:0]
  g0.w = (unsigned)((ga >> 32) & 0x01ffffffu)           // global_addr[56:32]
       | 0x80000000u;                                   // type=2 ("image")
  const i32x8 g1 = { 0x00020000,   // data_size=2 (4B), wg_mask=0
                     0x04000000,   // tensor_dim0 = 1024 (bits 79:48 -> <<16)
                     0x01000000,   // tensor_dim1 = 256  (bits 111:80 -> <<16)
                     0x00100000,   // tile_dim0   = 16   (bits 127:112)
                     0x00000100,   // tile_dim1   = 256  (bits 143:128)
                     0x00000400,   // tensor_dim0_stride = 1024 (bits 207:160)
                     0, 0 };       // tensor_dim1_stride = 0 (2D)
  const i32x4 gz = { 0, 0, 0, 0 };
#if __clang_major__ >= 23
  const i32x8 gz8 = { 0, 0, 0, 0, 0, 0, 0, 0 };
  __builtin_amdgcn_tensor_load_to_lds(g0, g1, gz, gz, gz8, 0);
#else
  __builtin_amdgcn_tensor_load_to_lds(g0, g1, gz, gz, 0);
#endif
}
#endif

// ---------------------------------------------------------------------------
// Kernel 2: fused GEMM (fp32 WMMA, dual acc chains) + argmin + gather
//   score[b,k] = e2[k] - 2 * x[b,:].W[:,k]
// ---------------------------------------------------------------------------
__global__ __launch_bounds__(256)
void vq_main_kernel(const float* __restrict__ x, const float* __restrict__ w,
                    const float* __restrict__ e2, float* __restrict__ out) {
  __shared__ float panel[2][D_DIM * 16];   // 2 x 16 KB: double-buffered W panel
  __shared__ float rvals[NWAVES][8][32];   // 8 KB: per-slot best values
  __shared__ int   ridx [NWAVES][8][32];   // 8 KB: per-slot best indices

  const int t    = threadIdx.x;
  const int lane = t & 31;
  const int wave = t >> 5;
  const int half = lane >> 4;   // 0: K=0,1 / 1: K=2,3 (A layout); C: M base 0/8
  const int l15  = lane & 15;

  const int rowBase = blockIdx.x * ROWS_PER_WG + wave * 16;
  const int myRow   = rowBase + l15;

  // --- Preload A fragments for this wave's 16-row stripe, scaled by -2 ---
  // ISA 16x4 f32 A layout: lanes 0-15 hold (K=0,K=1), lanes 16-31 hold (K=2,K=3)
  v2f afrag[64];
  const float* xrow = x + (size_t)myRow * D_DIM;
  #pragma unroll
  for (int s = 0; s < 64; ++s) {
    v2f a = *(const v2f*)(xrow + s * 4 + 2 * half);
    afrag[s] = a * -2.0f;
  }

  float bestVal[8];
  int   bestIdx[8];
  #pragma unroll
  for (int j = 0; j < 8; ++j) { bestVal[j] = 3.4e38f; bestIdx[j] = 0; }

#ifdef USE_TDM
  // Kick off DMA of tile 0 into buffer 0 (wave 0 only; TDM ignores EXEC,
  // so gate at wave granularity).
  if (wave == 0) {
    tdm_load_panel(w, (unsigned)(uintptr_t)&panel[0][0]);
  }
#endif

  for (int tile = 0; tile < NTILES; ++tile) {
    const int n0  = tile * 16;
    const int buf = tile & 1;

#ifdef USE_TDM
    __syncthreads();              // all waves done reading panel[buf^1]
    if (wave == 0) {
      if (tile + 1 < NTILES) {
        // DMA next panel into the buffer that was just released.
        tdm_load_panel(w + (size_t)(n0 + 16),
                       (unsigned)(uintptr_t)&panel[buf ^ 1][0]);
        __builtin_amdgcn_s_wait_tensorcnt(1);   // current tile's DMA done
      } else {
        __builtin_amdgcn_s_wait_tensorcnt(0);
      }
    }
    __syncthreads();              // publish panel[buf] to all waves
#else
    __syncthreads();              // previous tile's panel reads done
    #pragma unroll
    for (int i = 0; i < 16; ++i) {
      const int e = i * 256 + t;
      const int d = e >> 4, j = e & 15;
      panel[buf][e] = w[(size_t)d * K_DIM + n0 + j];
    }
    __syncthreads();
    if (tile + 1 < NTILES) {
      __builtin_prefetch(&w[(size_t)t * K_DIM + n0 + 16], 0, 0);
    }
#endif

    const float* pnl = &panel[buf][0];
    const float  e2v = e2[n0 + l15];   // overlaps with the WMMA loop below

    // Two independent accumulation chains to fill the XDL pipe.
    v8f acc0 = {};
    v8f acc1 = {};
    #pragma unroll
    for (int s = 0; s < 64; s += 2) {
      const int d0 = s * 4 + 2 * half;         // B 4x16 layout mirrors A
      const int d1 = (s + 1) * 4 + 2 * half;
      v2f b0, b1;
      b0.x = pnl[d0 * 16 + l15];
      b0.y = pnl[(d0 + 1) * 16 + l15];
      b1.x = pnl[d1 * 16 + l15];
      b1.y = pnl[(d1 + 1) * 16 + l15];
      acc0 = __builtin_amdgcn_wmma_f32_16x16x4_f32(
          false, afrag[s],     false, b0, (short)0, acc0, false, false);
      acc1 = __builtin_amdgcn_wmma_f32_16x16x4_f32(
          false, afrag[s + 1], false, b1, (short)0, acc1, false, false);
    }

    // Running per-slot argmin. C layout: vgpr j, lanes 0-15 -> (M=j, N=l15),
    // lanes 16-31 -> (M=8+j, N=l15). Candidate n strictly increases per slot,
    // so '<' keeps the first (lowest) index on exact ties.
    #pragma unroll
    for (int j = 0; j < 8; ++j) {
      const float v = acc0[j] + acc1[j] + e2v;
      if (v < bestVal[j]) { bestVal[j] = v; bestIdx[j] = n0 + l15; }
    }
  }

  // --- Cross-lane reduce: row M=r candidates live in slot j=r&7 across the
  //     16 lanes of half h=r>>3. ---
  #pragma unroll
  for (int j = 0; j < 8; ++j) {
    rvals[wave][j][lane] = bestVal[j];
    ridx [wave][j][lane] = bestIdx[j];
  }
  __syncthreads();

  int myIdx = 0;
  if (lane < 16) {
    const int j = lane & 7, h = lane >> 3;
    float bv = 3.5e38f;
    int   bi = 0x7fffffff;
    #pragma unroll
    for (int c = 0; c < 16; ++c) {
      const float v  = rvals[wave][j][h * 16 + c];
      const int   ix = ridx [wave][j][h * 16 + c];
      if (v < bv || (v == bv && ix < bi)) { bv = v; bi = ix; }
    }
    myIdx = bi;
  }

  // --- Gather: out[row,:] = W[:, argmin]; stores coalesced, reads are
  //     L2-resident columns of W (reused ~B/K = 64x each). ---
  for (int r = 0; r < 16; ++r) {
    const int idx = __shfl(myIdx, r, 32);
    float* orow = out + (size_t)(rowBase + r) * D_DIM;
    for (int d = lane; d < D_DIM; d += 32) {
      orow[d] = w[(size_t)d * K_DIM + idx];
    }
  }
}

// ---------------------------------------------------------------------------
extern "C" void kernel_launch(void* const* d_in, const int* in_sizes, int n_in,
                              void* d_out, int out_size, void* d_ws, size_t ws_size,
                              hipStream_t stream) {
  const float* x = (const float*)d_in[0];   // (B, 256)
  const float* w = (const float*)d_in[1];   // (256, 1024)
  float* out = (float*)d_out;               // (B, 256)
  float* e2  = (float*)d_ws;                // 1024 floats scratch

  const int B = in_sizes[0] / D_DIM;        // 65536

  vq_e2_kernel<<<K_DIM / 64, 256, 0, stream>>>(w, e2);
  vq_main_kernel<<<B / ROWS_PER_WG, 256, 0, stream>>>(x, w, e2, out);
}